// RNNAE_39737037423066
// MI455X (gfx1250) — compile-verified
//
#include <hip/hip_runtime.h>

typedef float v2f __attribute__((ext_vector_type(2)));
typedef float v8f __attribute__((ext_vector_type(8)));

// D = A(16x4 f32) * B(4x16 f32) + C(16x16 f32), full fp32 WMMA
__device__ __forceinline__ v8f wmma4(v2f a, v2f b, v8f c) {
  return __builtin_amdgcn_wmma_f32_16x16x4_f32(false, a, false, b, (short)0, c, false, false);
}

// lane <-> lane^16 swap, pure VALU (v_permlanex16_b32 with identity selectors)
__device__ __forceinline__ float swapx16(float x) {
  unsigned u = __builtin_bit_cast(unsigned, x);
  unsigned r = __builtin_amdgcn_permlanex16(u, u, 0x76543210u, 0xfedcba98u, false, false);
  return __builtin_bit_cast(float, r);
}

__device__ __forceinline__ float fsel(bool c, float a, float b) { return c ? a : b; }

// sigmoid(x) = rcp(1+exp2(-x*log2e)); tanh(x) = 2*rcp(1+exp2(-2x*log2e)) - 1
// tL/tH select tanh (true) vs sigmoid (false) for low/high lane halves (compile-time).
__device__ __forceinline__ float act_gate(float x, bool L, bool tL, bool tH) {
  const float SS = -1.4426950408889634f;
  const float ST = -2.8853900817779268f;
  float S = (tL == tH) ? (tL ? ST : SS) : fsel(L, tL ? ST : SS, tH ? ST : SS);
  float A = (tL == tH) ? (tL ? 2.f : 1.f) : fsel(L, tL ? 2.f : 1.f, tH ? 2.f : 1.f);
  float D = (tL == tH) ? (tL ? -1.f : 0.f) : fsel(L, tL ? -1.f : 0.f, tH ? -1.f : 0.f);
  float e = __builtin_amdgcn_exp2f(x * S);
  float r = __builtin_amdgcn_rcpf(1.f + e);
  return __builtin_fmaf(A, r, D);
}

#define N_STEPS 10

__global__ void __launch_bounds__(256)
rnn_rollout_wmma(const float* __restrict__ s_star, const int* __restrict__ go_mask,
                 const float* __restrict__ W_ih, const float* __restrict__ W_hh,
                 const float* __restrict__ b_ih, const float* __restrict__ b_hh,
                 const float* __restrict__ W1, const float* __restrict__ b1,
                 const float* __restrict__ W2, const float* __restrict__ b2,
                 const float* __restrict__ W3, const float* __restrict__ b3,
                 float* __restrict__ out, int ntiles) {
  const int lane = threadIdx.x & 31;
  const int m = lane & 15;       // batch row (B-frag) / out-feature row (A-frag)
  const int hi = lane >> 4;      // which wave half
  const bool L = (hi == 0);

  // ================= weight fragments (A operands, M=out-feat, K=in-feat) =================
  // Gates: Wcat(40x12) = [W_ih(40x2) | W_hh(40x10)], 3 M-tiles x 3 K-chunks
  v2f gA[3][3];
#pragma unroll
  for (int t = 0; t < 3; ++t)
#pragma unroll
    for (int c = 0; c < 3; ++c) {
      const int row = 16 * t + m;
      const int k0 = 4 * c + 2 * hi;
      float x = 0.f, y = 0.f;
      if (row < 40) {
        x = (k0 == 0) ? W_ih[row * 2 + 0]
                      : (k0 == 1) ? W_ih[row * 2 + 1] : W_hh[row * 10 + (k0 - 2)];
        const int k1 = k0 + 1;
        y = (k1 == 1) ? W_ih[row * 2 + 1] : W_hh[row * 10 + (k1 - 2)];
      }
      gA[t][c] = (v2f){x, y};
    }
  // Gate bias as C-operand fragments (D layout: VGPR r -> feat 16t+r (low) / 16t+r+8 (high))
  v8f gB[3];
#pragma unroll
  for (int t = 0; t < 3; ++t)
#pragma unroll
    for (int r = 0; r < 8; ++r) {
      const int f = 16 * t + r + 8 * hi;
      gB[t][r] = (f < 40) ? (b_ih[f] + b_hh[f]) : 0.f;
    }
  // W1 (20x10): 2 M-tiles x 3 K-chunks (K padded 10->12)
  v2f A1[2][3];
#pragma unroll
  for (int t = 0; t < 2; ++t)
#pragma unroll
    for (int c = 0; c < 3; ++c) {
      const int row = 16 * t + m, k0 = 4 * c + 2 * hi;
      float x = (row < 20 && k0 < 10) ? W1[row * 10 + k0] : 0.f;
      float y = (row < 20 && k0 + 1 < 10) ? W1[row * 10 + k0 + 1] : 0.f;
      A1[t][c] = (v2f){x, y};
    }
  v8f B1f[2];
#pragma unroll
  for (int t = 0; t < 2; ++t)
#pragma unroll
    for (int r = 0; r < 8; ++r) {
      const int f = 16 * t + r + 8 * hi;
      B1f[t][r] = (f < 20) ? b1[f] : 0.f;
    }
  // W2 (20x20): 2 M-tiles x 5 K-chunks
  v2f A2[2][5];
#pragma unroll
  for (int t = 0; t < 2; ++t)
#pragma unroll
    for (int c = 0; c < 5; ++c) {
      const int row = 16 * t + m, k0 = 4 * c + 2 * hi;
      float x = (row < 20) ? W2[row * 20 + k0] : 0.f;
      float y = (row < 20) ? W2[row * 20 + k0 + 1] : 0.f;
      A2[t][c] = (v2f){x, y};
    }
  v8f B2f[2];
#pragma unroll
  for (int t = 0; t < 2; ++t)
#pragma unroll
    for (int r = 0; r < 8; ++r) {
      const int f = 16 * t + r + 8 * hi;
      B2f[t][r] = (f < 20) ? b2[f] : 0.f;
    }
  // W3 (1x20): 1 M-tile (row 0 only) x 5 K-chunks
  v2f A3[5];
#pragma unroll
  for (int c = 0; c < 5; ++c) {
    const int k0 = 4 * c + 2 * hi;
    float x = (m == 0) ? W3[k0] : 0.f;
    float y = (m == 0) ? W3[k0 + 1] : 0.f;
    A3[c] = (v2f){x, y};
  }
  const float b3v = b3[0];

  float coef[N_STEPS];
#pragma unroll
  for (int t = 0; t < N_STEPS; ++t)
    coef[t] = (go_mask[t] != 0) ? ((t < 4) ? 5.0f : 2.5f) : 0.0f;

  // ================= per-tile rollout (16 rollouts per wave, grid-stride) =================
  const int wslot = blockIdx.x * (blockDim.x >> 5) + (threadIdx.x >> 5);
  const int nw = gridDim.x * (blockDim.x >> 5);
  for (int tile = wslot; tile < ntiles; tile += nw) {
    const int base = tile * 16;
    const float sst = s_star[base + m];  // replicated in both halves
    float s = 0.f, err = 0.f;
    v8f hd = {};   // h in D layout: low lanes feat r, high lanes feat r+8 (r=0,1; rest zero)
    v8f cc = {};   // c, same layout
    float hsw[8];
#pragma unroll
    for (int r = 0; r < 8; ++r) hsw[r] = 0.f;

#pragma unroll
    for (int t = 0; t < N_STEPS; ++t) {
      const float ah = coef[t] * (sst - 0.6f);

      // ---- gates: Wcat @ [s; ah; h]  (B-frags: Z^T columns = batch) ----
      v2f q0 = (v2f){fsel(L, s, hsw[0]), fsel(L, ah, hsw[1])};
      v2f q1 = (v2f){fsel(L, hd[2], hsw[4]), fsel(L, hd[3], hsw[5])};
      v2f q2 = (v2f){fsel(L, hd[6], hd[0]), fsel(L, hd[7], hd[1])};
      v8f g0 = gB[0], g1 = gB[1], g2 = gB[2];
      g0 = wmma4(gA[0][0], q0, g0); g0 = wmma4(gA[0][1], q1, g0); g0 = wmma4(gA[0][2], q2, g0);
      g1 = wmma4(gA[1][0], q0, g1); g1 = wmma4(gA[1][1], q1, g1); g1 = wmma4(gA[1][2], q2, g1);
      g2 = wmma4(gA[2][0], q0, g2); g2 = wmma4(gA[2][1], q1, g2); g2 = wmma4(gA[2][2], q2, g2);

      // ---- nonlinearities (feats: i 0-9 sig, f 10-19 sig, g 20-29 tanh, o 30-39 sig) ----
      v8f a0v, a1v, a2v;
#pragma unroll
      for (int r = 0; r < 8; ++r) a0v[r] = act_gate(g0[r], L, false, false);
#pragma unroll
      for (int r = 0; r < 4; ++r) a1v[r] = act_gate(g1[r], L, false, true);
      a1v[4] = act_gate(g1[4], L, true, true);
      a1v[5] = act_gate(g1[5], L, true, true);
      a1v[6] = act_gate(g1[6], L, true, false);
      a1v[7] = act_gate(g1[7], L, true, false);
#pragma unroll
      for (int r = 0; r < 8; ++r) a2v[r] = act_gate(g2[r], L, false, false);

      // ---- realign f/g/o tiles onto i's (slot = hidden unit j) layout ----
      const float s0_2 = swapx16(a0v[2]), s0_3 = swapx16(a0v[3]), s0_4 = swapx16(a0v[4]);
      const float s0_5 = swapx16(a0v[5]), s0_6 = swapx16(a0v[6]), s0_7 = swapx16(a0v[7]);
      const float s1_0 = swapx16(a1v[0]), s1_1 = swapx16(a1v[1]), s1_2 = swapx16(a1v[2]);
      const float s1_3 = swapx16(a1v[3]), s1_6 = swapx16(a1v[6]), s1_7 = swapx16(a1v[7]);
      const float s2_6 = swapx16(a2v[6]), s2_7 = swapx16(a2v[7]);
      float fv[8], gv[8], ov[8];
      fv[0] = fsel(L, s0_2, s1_2); fv[1] = fsel(L, s0_3, s1_3);
      fv[2] = s0_4; fv[3] = s0_5; fv[4] = s0_6; fv[5] = s0_7;
      fv[6] = a1v[0]; fv[7] = a1v[1];
      gv[0] = a1v[4]; gv[1] = a1v[5]; gv[2] = a1v[6]; gv[3] = a1v[7];
      gv[4] = s1_0; gv[5] = s1_1; gv[6] = s1_2; gv[7] = s1_3;
      ov[0] = fsel(L, s1_6, s2_6); ov[1] = fsel(L, s1_7, s2_7);
      ov[2] = a2v[0]; ov[3] = a2v[1]; ov[4] = a2v[2];
      ov[5] = a2v[3]; ov[6] = a2v[4]; ov[7] = a2v[5];

      // ---- c = f*c + i*g ; h = o*tanh(c) ----
#pragma unroll
      for (int r = 0; r < 8; ++r) {
        cc[r] = __builtin_fmaf(fv[r], cc[r], a0v[r] * gv[r]);
        hd[r] = ov[r] * act_gate(cc[r], L, true, true);
      }
#pragma unroll
      for (int r = 2; r < 8; ++r) {  // dead slots (feats 10-15 in high half)
        hd[r] = fsel(L, hd[r], 0.f);
        cc[r] = fsel(L, cc[r], 0.f);
      }
#pragma unroll
      for (int r = 0; r < 8; ++r) hsw[r] = swapx16(hd[r]);

      // ---- MLP1: relu(W1 @ h + b1), K=10 (chunks 0-2, last half zero) ----
      v2f m0 = (v2f){fsel(L, hd[0], hsw[2]), fsel(L, hd[1], hsw[3])};
      v2f m1 = (v2f){fsel(L, hd[4], hsw[6]), fsel(L, hd[5], hsw[7])};
      v2f m2 = (v2f){fsel(L, hsw[0], 0.f), fsel(L, hsw[1], 0.f)};
      v8f u0 = B1f[0], u1 = B1f[1];
      u0 = wmma4(A1[0][0], m0, u0); u0 = wmma4(A1[0][1], m1, u0); u0 = wmma4(A1[0][2], m2, u0);
      u1 = wmma4(A1[1][0], m0, u1); u1 = wmma4(A1[1][1], m1, u1); u1 = wmma4(A1[1][2], m2, u1);
#pragma unroll
      for (int r = 0; r < 8; ++r) u0[r] = fmaxf(u0[r], 0.f);
#pragma unroll
      for (int r = 0; r < 4; ++r) u1[r] = fsel(L, fmaxf(u1[r], 0.f), 0.f);
#pragma unroll
      for (int r = 4; r < 8; ++r) u1[r] = 0.f;

      // ---- MLP2: relu(W2 @ h1 + b2), K=20 (5 chunks) ----
      float p0[8];
#pragma unroll
      for (int r = 0; r < 8; ++r) p0[r] = swapx16(u0[r]);
      const float p1_2 = swapx16(u1[2]), p1_3 = swapx16(u1[3]);
      v2f f2[5];
      f2[0] = (v2f){fsel(L, u0[0], p0[2]), fsel(L, u0[1], p0[3])};
      f2[1] = (v2f){fsel(L, u0[4], p0[6]), fsel(L, u0[5], p0[7])};
      f2[2] = (v2f){fsel(L, p0[0], u0[2]), fsel(L, p0[1], u0[3])};
      f2[3] = (v2f){fsel(L, p0[4], u0[6]), fsel(L, p0[5], u0[7])};
      f2[4] = (v2f){fsel(L, u1[0], p1_2), fsel(L, u1[1], p1_3)};
      v8f v0 = B2f[0], v1 = B2f[1];
#pragma unroll
      for (int c = 0; c < 5; ++c) { v0 = wmma4(A2[0][c], f2[c], v0); v1 = wmma4(A2[1][c], f2[c], v1); }
#pragma unroll
      for (int r = 0; r < 8; ++r) v0[r] = fmaxf(v0[r], 0.f);
#pragma unroll
      for (int r = 0; r < 4; ++r) v1[r] = fsel(L, fmaxf(v1[r], 0.f), 0.f);
#pragma unroll
      for (int r = 4; r < 8; ++r) v1[r] = 0.f;

      // ---- MLP3: ar = W3 @ h2 + b3 ----
      float z0[8];
#pragma unroll
      for (int r = 0; r < 8; ++r) z0[r] = swapx16(v0[r]);
      const float z1_2 = swapx16(v1[2]), z1_3 = swapx16(v1[3]);
      v2f f3[5];
      f3[0] = (v2f){fsel(L, v0[0], z0[2]), fsel(L, v0[1], z0[3])};
      f3[1] = (v2f){fsel(L, v0[4], z0[6]), fsel(L, v0[5], z0[7])};
      f3[2] = (v2f){fsel(L, z0[0], v0[2]), fsel(L, z0[1], v0[3])};
      f3[3] = (v2f){fsel(L, z0[4], v0[6]), fsel(L, z0[5], v0[7])};
      f3[4] = (v2f){fsel(L, v1[0], z1_2), fsel(L, v1[1], z1_3)};
      v8f d3 = {};
#pragma unroll
      for (int c = 0; c < 5; ++c) d3 = wmma4(A3[c], f3[c], d3);
      const float ar = d3[0];                      // row M=0: ar for batch=lane (low half)
      const float arAll = fsel(L, ar, swapx16(ar)); // replicate into high half
      s += arAll + b3v;
      const float df = s - sst;
      err = __builtin_fmaf(df, df, err);
    }
    out[base + m] = err;  // both halves write identical values
  }
}

extern "C" void kernel_launch(void* const* d_in, const int* in_sizes, int n_in,
                              void* d_out, int out_size, void* d_ws, size_t ws_size,
                              hipStream_t stream) {
  (void)n_in; (void)d_ws; (void)ws_size; (void)out_size;
  const float* s_star = (const float*)d_in[0];
  const int*   go     = (const int*)d_in[1];
  const float* W_ih   = (const float*)d_in[2];
  const float* W_hh   = (const float*)d_in[3];
  const float* b_ih   = (const float*)d_in[4];
  const float* b_hh   = (const float*)d_in[5];
  const float* W1     = (const float*)d_in[6];
  const float* b1     = (const float*)d_in[7];
  const float* W2     = (const float*)d_in[8];
  const float* b2     = (const float*)d_in[9];
  const float* W3     = (const float*)d_in[10];
  const float* b3     = (const float*)d_in[11];
  float* out = (float*)d_out;

  const int ntiles = in_sizes[0] >> 4;           // 16 rollouts per wave-tile
  int waves = (ntiles + 7) / 8;                  // ~8 tiles per wave
  int blocks = (waves + 7) / 8;                  // 8 waves (256 threads) per block
  if (blocks < 1) blocks = 1;
  if (blocks > 16384) blocks = 16384;
  rnn_rollout_wmma<<<dim3(blocks), dim3(256), 0, stream>>>(
      s_star, go, W_ih, W_hh, b_ih, b_hh, W1, b1, W2, b2, W3, b3, out, ntiles);
}